// Muscles_SRT_8581344658236
// MI455X (gfx1250) — compile-verified
//
#include <hip/hip_runtime.h>
#include <math.h>

#define ATAN5_F 1.3734007669450157f   // atan(5.0)

// One thread handles one (b, j) pair => both muscles m=0,1.
//   muscle_SS[b,j,:,:] is a contiguous float4; lMT = (.x, .y)
//   alphas[b,j,:]      is a contiguous float2
// Outputs: d_out = [ F (B*J*2) | Kout (B*J*2) ] flat fp32.
__global__ __launch_bounds__(256) void muscles_srt_kernel(
    const float* __restrict__ alphas,      // B*J*2
    const float* __restrict__ muscle_SS,   // B*J*2*2
    const float* __restrict__ fMax_raw,    // J*2
    const float* __restrict__ lM_opt_raw,  // J*2
    const float* __restrict__ c_raw,       // J*2
    const float* __restrict__ A_raw,       // J*2
    float* __restrict__ out,               // 2 * B*J*2
    int total)                             // B*J
{
    const int t = blockIdx.x * blockDim.x + threadIdx.x;
    if (t >= total) return;
    const int j = t & 15;                  // J == 16

    // gfx1250 prefetch: stream ~64KB ahead of the heavy compute loop.
    // locality=3 -> WGP-scope prefetch (pulls into all cache levels),
    // so the line is near when the b128/b64 loads of the next blocks hit.
    __builtin_prefetch(&((const float4*)muscle_SS)[t + 4096], 0, 3);
    __builtin_prefetch(&((const float2*)alphas)[t + 8192], 0, 3);

    const float4 ss = ((const float4*)muscle_SS)[t];   // global_load_b128
    const float2 al = ((const float2*)alphas)[t];      // global_load_b64
    const float2 fMr = ((const float2*)fMax_raw)[j];
    const float2 lOr = ((const float2*)lM_opt_raw)[j];
    const float2 cr  = ((const float2*)c_raw)[j];
    const float2 Ar  = ((const float2*)A_raw)[j];

    float a[2], lMT[2], fMax[2], lM_opt[2], c[2], A[2];
    a[0] = fminf(fmaxf(al.x, 0.01f), 1.0f);
    a[1] = fminf(fmaxf(al.y, 0.01f), 1.0f);
    lMT[0] = ss.x; lMT[1] = ss.y;
    fMax[0]   = __expf(fMr.x); fMax[1]   = __expf(fMr.y);
    lM_opt[0] = __expf(lOr.x); lM_opt[1] = __expf(lOr.y);
    c[0] = __expf(cr.x); c[1] = __expf(cr.y);
    A[0] = __expf(Ar.x); A[1] = __expf(Ar.y);

    float epsMT[2], epsM[2], velM[2];
    #pragma unroll
    for (int m = 0; m < 2; ++m) {
        epsMT[m] = lMT[m] / lM_opt[m];
        epsM[m] = 0.0f;
        velM[m] = 0.0f;
    }

    const float INF = __builtin_inff();

    // 10 fixed Newton iterations; the two muscles are independent chains
    // (unrolled pair) so VALU + TRANS32 (v_exp_f32 / v_rcp) can co-execute.
    for (int it = 0; it < 10; ++it) {
        #pragma unroll
        for (int m = 0; m < 2; ++m) {
            const float am = a[m];
            const float d  = epsMT[m] - epsM[m];
            const float gt = (d > 0.0f) ? 1.0f : 0.0f;
            const float e_neg = __expf(-100.0f * d);          // v_exp_f32

            const float eM  = epsM[m];
            const float flq = 1.0f - 4.0f * eM * eM;
            const float fl  = fmaxf(flq, 0.0f);
            const float fl_pos = (flq > 0.0f) ? 1.0f : 0.0f;

            const float v   = velM[m];
            const float fv  = 1.0f - atanf(0.5f * v) * (1.0f / ATAN5_F);
            const float fv_rel = fmaxf(fv, 0.0f);
            const bool  vok = (v >= -10.0f);
            const float vmask = vok ? 1.0f : 0.0f;
            const float dv  = 0.25f * v * v + 1.0f;

            const float rdv = 1.0f / (ATAN5_F * dv);          // shared reciprocal

            const float g0 = -10.0f * e_neg * gt
                             - 8.0f * am * eM * fl_pos * fv_rel;
            const float g1 = -0.5f * am * fl * rdv * vmask;

            const float h00 = 1000.0f * e_neg * gt - 8.0f * am * fv_rel + 0.0001f;
            // reference: num / (ATAN5*dv*vmask). vmask in {0,1}:
            //   vmask==1 -> exact; vmask==0 -> IEEE x/0 == x*INF (inf/-inf/nan)
            const float h01n = 4.0f * am * eM * fl_pos * rdv;
            const float h01  = vok ? h01n : h01n * INF;
            const float h11 = 0.25f * am * v * vmask * fl * (rdv * rdv * ATAN5_F)
                              + 0.0001f;

            const float det    = h00 * h11 - h01 * h01;
            const float invdet = 1.0f / det;
            const float u0 = -(h11 * g0 - h01 * g1) * invdet;
            const float u1 = -(h00 * g1 - h01 * g0) * invdet;
            epsM[m] = eM + u0;
            velM[m] = v + u1;
        }
    }

    float F[2], K[2];
    #pragma unroll
    for (int m = 0; m < 2; ++m) {
        const float eMT  = epsMT[m];
        const float eMT2 = eMT * eMT;
        const float expc = __expf(c[m] * eMT2);
        const float fPEr = 2.0f * c[m] * A[m] * eMT * expc;
        const float fPE  = fmaxf(fPEr, 0.0f);
        const float expSE = __expf(100.0f * (eMT - epsM[m]));
        const float fSEr  = 0.1f * (expSE - 1.0f);
        const float fSE   = fmaxf(fSEr, 0.0f);

        F[m] = fMax[m] * (fSE + fPE);

        const float se_pos = (fSEr > 0.0f) ? 1.0f : 0.0f;
        const float pe_pos = (fPEr > 0.0f) ? 1.0f : 0.0f;
        K[m] = fMax[m] / lM_opt[m] *
               (10.0f * expSE * se_pos +
                (2.0f * c[m] * A[m] * expc +
                 4.0f * c[m] * c[m] * A[m] * eMT2 * expc * pe_pos));
    }

    float2 Fo; Fo.x = F[0]; Fo.y = F[1];
    float2 Ko; Ko.x = K[0]; Ko.y = K[1];
    ((float2*)out)[t] = Fo;                          // F block
    ((float2*)(out + 2 * (size_t)total))[t] = Ko;    // Kout block
}

extern "C" void kernel_launch(void* const* d_in, const int* in_sizes, int n_in,
                              void* d_out, int out_size, void* d_ws, size_t ws_size,
                              hipStream_t stream) {
    const float* alphas     = (const float*)d_in[0];
    const float* muscle_SS  = (const float*)d_in[1];
    const float* fMax_raw   = (const float*)d_in[2];
    const float* lM_opt_raw = (const float*)d_in[3];
    const float* c_raw      = (const float*)d_in[4];
    const float* A_raw      = (const float*)d_in[5];

    const int total = in_sizes[0] / 2;   // B*J (alphas has B*J*2 elements)
    const int threads = 256;             // 8 wave32 waves per block
    const int blocks = (total + threads - 1) / threads;

    muscles_srt_kernel<<<blocks, threads, 0, stream>>>(
        alphas, muscle_SS, fMax_raw, lM_opt_raw, c_raw, A_raw,
        (float*)d_out, total);
}